// TripletLoss_41953240547465
// MI455X (gfx1250) — compile-verified
//
#include <hip/hip_runtime.h>
#include <hip/hip_bf16.h>

// CDNA5 (gfx1250) wave32 WMMA types
typedef __attribute__((ext_vector_type(2))) float v2f;   // A/B fragment: 16x4 / 4x16 f32
typedef __attribute__((ext_vector_type(8))) float v8f;   // C/D fragment: 16x16 f32

#define BS 512
#define MARGIN 0.5f
#define EPS_NORM 1e-10f

// ---------------------------------------------------------------------------
// Kernel A: per-row sum of squares -> inv_norm[i] = 1/max(||row||, eps),
//           sq[i] = sum(p_hat_i^2) = ss * inv^2  (matches reference exactly)
// ---------------------------------------------------------------------------
__global__ void row_norms_kernel(const float* __restrict__ pred,
                                 float* __restrict__ inv_norm,
                                 float* __restrict__ sq) {
    __shared__ float red[256];
    const int row = blockIdx.x;
    const int tid = threadIdx.x;
    const float* p = pred + row * BS;
    float s = 0.0f;
    for (int c = tid; c < BS; c += 256) {
        float v = p[c];
        s += v * v;
    }
    red[tid] = s;
    __syncthreads();
    for (int off = 128; off > 0; off >>= 1) {
        if (tid < off) red[tid] += red[tid + off];
        __syncthreads();
    }
    if (tid == 0) {
        float ss  = red[0];
        float iv  = 1.0f / fmaxf(sqrtf(ss), EPS_NORM);
        inv_norm[row] = iv;
        sq[row]       = ss * iv * iv;
    }
}

// ---------------------------------------------------------------------------
// Kernel B: Gram matrix via V_WMMA_F32_16X16X4_F32, fused distance epilogue.
// One wave (32 lanes) computes one 16x16 tile of dist. 32x32 tiles = 1024
// waves; 8 waves/block -> 128 blocks.
//
// A 16x4 f32 layout (ISA 7.12.2): lane = half*16 + m holds A[m][2*half],
// A[m][2*half+1]  -> contiguous float2 from pred row (rowBase+m).
// B 4x16 layout is symmetric over the column tile's rows (P^T).
// C/D: VGPR r -> row (rowBase + r + 8*half), col (colBase + m).
// ---------------------------------------------------------------------------
__global__ void gram_dist_kernel(const float* __restrict__ pred,
                                 const float* __restrict__ inv_norm,
                                 const float* __restrict__ sq,
                                 float* __restrict__ dist) {
    const int wave = threadIdx.x >> 5;
    const int lane = threadIdx.x & 31;
    const int tile = blockIdx.x * 8 + wave;          // 0..1023
    const int tileM = tile >> 5;                      // 0..31
    const int tileN = tile & 31;                      // 0..31
    const int rowBase = tileM << 4;
    const int colBase = tileN << 4;
    const int half = lane >> 4;                       // 0 or 1
    const int m    = lane & 15;

    const int aRow = rowBase + m;
    const int bRow = colBase + m;
    const float ai = inv_norm[aRow];
    const float bi = inv_norm[bRow];
    const float* aPtr = pred + aRow * BS + 2 * half;
    const float* bPtr = pred + bRow * BS + 2 * half;

    v8f c = {};
#pragma unroll 8
    for (int k = 0; k < BS; k += 4) {
        v2f a, b;
        a.x = aPtr[k]     * ai;
        a.y = aPtr[k + 1] * ai;
        b.x = bPtr[k]     * bi;
        b.y = bPtr[k + 1] * bi;
        // D = A x B + C, f32 in / f32 out (emits v_wmma_f32_16x16x4_f32)
        c = __builtin_amdgcn_wmma_f32_16x16x4_f32(
                /*neg_a=*/false, a, /*neg_b=*/false, b,
                /*c_mod=*/(short)0, c, /*reuse_a=*/false, /*reuse_b=*/false);
    }

    const int col = colBase + m;
    const float sqc = sq[col];
#pragma unroll
    for (int r = 0; r < 8; ++r) {
        const int row = rowBase + r + 8 * half;
        float d2 = sq[row] + sqc - 2.0f * c[r];
        d2 = fmaxf(d2, 0.0f);
        dist[row * BS + col] = (d2 > 0.0f) ? sqrtf(d2) : 0.0f;
    }
}

// ---------------------------------------------------------------------------
// Kernel C: per-anchor triplet reduction. Block = anchor i (512 blocks, 256
// threads). Valid triplet: t[i]==1, t[j]==1, j!=i, t[k]==0.
// contribution = relu(dist[i,j] - dist[i,k] + MARGIN); also count of >0.
// Row i of dist + targets staged in LDS; invalid j rows skipped wholesale.
// ---------------------------------------------------------------------------
__global__ void triplet_partial_kernel(const float* __restrict__ dist,
                                       const int* __restrict__ target,
                                       float* __restrict__ psum,
                                       float* __restrict__ pcnt) {
    __shared__ float drow[BS];
    __shared__ unsigned char tg[BS];
    __shared__ float rs[256];
    __shared__ float rc[256];

    const int i   = blockIdx.x;
    const int tid = threadIdx.x;

    for (int c = tid; c < BS; c += 256) {
        drow[c] = dist[i * BS + c];
        tg[c]   = (unsigned char)target[c];
    }
    __syncthreads();

    float sum = 0.0f, cnt = 0.0f;
    if (tg[i] == 1) {
        for (int j = 0; j < BS; ++j) {
            if (tg[j] != 1 || j == i) continue;      // uniform across block
            const float dij = drow[j] + MARGIN;
#pragma unroll
            for (int k = tid; k < BS; k += 256) {    // 2 iterations/thread
                if (tg[k] == 0) {
                    float v = dij - drow[k];
                    if (v > 0.0f) { sum += v; cnt += 1.0f; }
                }
            }
        }
    }

    rs[tid] = sum;
    rc[tid] = cnt;
    __syncthreads();
    for (int off = 128; off > 0; off >>= 1) {
        if (tid < off) { rs[tid] += rs[tid + off]; rc[tid] += rc[tid + off]; }
        __syncthreads();
    }
    if (tid == 0) { psum[i] = rs[0]; pcnt[i] = rc[0]; }
}

// ---------------------------------------------------------------------------
// Kernel D: final reduction of 512 partials -> scalar loss.
// ---------------------------------------------------------------------------
__global__ void finalize_kernel(const float* __restrict__ psum,
                                const float* __restrict__ pcnt,
                                float* __restrict__ out) {
    __shared__ float rs[256];
    __shared__ float rc[256];
    const int tid = threadIdx.x;
    float s = 0.0f, c = 0.0f;
    for (int i = tid; i < BS; i += 256) { s += psum[i]; c += pcnt[i]; }
    rs[tid] = s;
    rc[tid] = c;
    __syncthreads();
    for (int off = 128; off > 0; off >>= 1) {
        if (tid < off) { rs[tid] += rs[tid + off]; rc[tid] += rc[tid + off]; }
        __syncthreads();
    }
    if (tid == 0) {
        // LOSS_WEIGHT == 1.0
        out[0] = rs[0] / (rc[0] + 1e-7f);
    }
}

// ---------------------------------------------------------------------------
// Launch: pred [512*512] f32, target [512] i32 -> out [1] f32.
// Workspace layout (floats): inv[512] | sq[512] | dist[512*512] | psum[512]
// | pcnt[512]  (~1.03 MB total).
// ---------------------------------------------------------------------------
extern "C" void kernel_launch(void* const* d_in, const int* in_sizes, int n_in,
                              void* d_out, int out_size, void* d_ws, size_t ws_size,
                              hipStream_t stream) {
    const float* pred   = (const float*)d_in[0];
    const int*   target = (const int*)d_in[1];
    float*       out    = (float*)d_out;

    float* w        = (float*)d_ws;
    float* inv_norm = w;                         // 512
    float* sq       = w + BS;                    // 512
    float* dist     = w + 2 * BS;                // 512*512
    float* psum     = dist + BS * BS;            // 512
    float* pcnt     = psum + BS;                 // 512

    row_norms_kernel<<<BS, 256, 0, stream>>>(pred, inv_norm, sq);
    gram_dist_kernel<<<128, 256, 0, stream>>>(pred, inv_norm, sq, dist);
    triplet_partial_kernel<<<BS, 256, 0, stream>>>(dist, target, psum, pcnt);
    finalize_kernel<<<1, 256, 0, stream>>>(psum, pcnt, out);
}